// HashEncoder_22857815949552
// MI455X (gfx1250) — compile-verified
//
#include <hip/hip_runtime.h>
#include <stdint.h>

// instant-ngp spatial hash primes (PRIME0 == 1)
#define PRIME1 2654435761u
#define PRIME2 805459861u

typedef float v2f __attribute__((ext_vector_type(2)));

// Raw 32-bit LDS offset of a shared-memory object (addrspace(3) ptrtoint).
__device__ __forceinline__ uint32_t lds_addr_of(const void* p) {
    return (uint32_t)(uintptr_t)(__attribute__((address_space(3))) const void*)p;
}

__global__ __launch_bounds__(256) void hashenc_fwd(
    const float* __restrict__ inputs,   // (B, 3)
    const float* __restrict__ emb,      // (7114752, 2)
    float* __restrict__ out,            // (B, 32)
    int npoints)
{
    // Block = 256 threads = 16 points x 16 levels. Stage the block's 48 input
    // floats into LDS via the CDNA5 async global->LDS path (ASYNCcnt).
    __shared__ float sxyz[48];
    const int t = threadIdx.x;
    const int fbase = blockIdx.x * 48;          // float index of block's coords
    const int nfloats = npoints * 3;

    if (t < 48 && (fbase + t) < nfloats) {
        uint32_t lds  = lds_addr_of(&sxyz[t]);
        uint32_t goff = (uint32_t)t * 4u;       // per-lane byte offset (VADDR)
        const float* src = inputs + fbase;      // uniform -> SGPR pair (SADDR)
        asm volatile("global_load_async_to_lds_b32 %0, %1, %2"
                     :: "v"(lds), "v"(goff), "s"(src) : "memory");
    }
    asm volatile("s_wait_asynccnt 0" ::: "memory");
    __syncthreads();

    const uint32_t gtid = blockIdx.x * 256u + (uint32_t)t;
    const uint32_t b = gtid >> 4;               // point index
    const uint32_t l = gtid & 15u;              // level index
    if (b >= (uint32_t)npoints) return;

    const int pl = t >> 4;                      // point-in-block
    const float x = (sxyz[pl * 3 + 0] + 1.0f) * 0.5f;   // map [-1,1) -> [0,1)
    const float y = (sxyz[pl * 3 + 1] + 1.0f) * 0.5f;
    const float z = (sxyz[pl * 3 + 2] + 1.0f) * 0.5f;

    // Per-level constants: res = 16<<l, scale = res-1 (exact in fp32).
    const uint32_t res = 16u << l;
    const float scale = (float)res - 1.0f;

    const float px = x * scale, py = y * scale, pz = z * scale;
    const float fx = floorf(px), fy = floorf(py), fz = floorf(pz);
    float wx = px - fx, wy = py - fy, wz = pz - fz;
    wx = wx * wx * (3.0f - 2.0f * wx);           // smoothstep weights
    wy = wy * wy * (3.0f - 2.0f * wy);
    wz = wz * wz * (3.0f - 2.0f * wz);
    const uint32_t x0 = (uint32_t)fx, y0 = (uint32_t)fy, z0 = (uint32_t)fz;

    // Level base offsets: [0,4096,36864,299008, 299008+(l-3)*524288...]
    const uint32_t offb = (l < 3u)
        ? ((l == 0u) ? 0u : (l == 1u) ? 4096u : 36864u)
        : 299008u + (l - 3u) * 524288u;

    // Incremental corner index terms (x-corner step for hash is *PRIME0==1).
    uint32_t ax0, ax1, ay0, ay1, az0, az1;
    const bool hashed = (l >= 3u);
    if (hashed) {
        ax0 = x0;           ax1 = x0 + 1u;
        ay0 = y0 * PRIME1;  ay1 = ay0 + PRIME1;
        az0 = z0 * PRIME2;  az1 = az0 + PRIME2;
    } else {
        const uint32_t r2 = res * res;
        ax0 = x0;           ax1 = x0 + 1u;
        ay0 = y0 * res;     ay1 = ay0 + res;
        az0 = z0 * r2;      az1 = az0 + r2;
    }

    // All 8 indices first, then all 8 gathers -> compiler batches the
    // global_load_b64s for max memory-level parallelism (L2-resident table).
    uint32_t idx[8];
#pragma unroll
    for (int k = 0; k < 8; ++k) {               // corner bits: x=k>>2,y=k>>1&1,z=k&1
        const uint32_t ax = (k & 4) ? ax1 : ax0;
        const uint32_t ay = (k & 2) ? ay1 : ay0;
        const uint32_t az = (k & 1) ? az1 : az0;
        idx[k] = hashed ? ((ax ^ ay ^ az) & 0x7FFFFu) : (ax + ay + az);
    }

    const v2f* __restrict__ emb2 = (const v2f*)emb;
    v2f v[8];
#pragma unroll
    for (int k = 0; k < 8; ++k)
        v[k] = emb2[offb + idx[k]];

    const float ox = 1.0f - wx, oy = 1.0f - wy, oz = 1.0f - wz;
    const float wyz[4] = { oy * oz, oy * wz, wy * oz, wy * wz };
    float acc0 = 0.0f, acc1 = 0.0f;
#pragma unroll
    for (int k = 0; k < 8; ++k) {
        const float wk = ((k & 4) ? wx : ox) * wyz[k & 3];
        acc0 = fmaf(wk, v[k].x, acc0);
        acc1 = fmaf(wk, v[k].y, acc1);
    }

    // Streamed 128MB output: non-temporal so it never evicts the ~57MB
    // embedding table from the 192MB L2. Lanes write consecutive float2s.
    v2f o; o.x = acc0; o.y = acc1;
    __builtin_nontemporal_store(o, (v2f*)out + (size_t)b * 16 + l);
}

extern "C" void kernel_launch(void* const* d_in, const int* in_sizes, int n_in,
                              void* d_out, int out_size, void* d_ws, size_t ws_size,
                              hipStream_t stream) {
    const float* inputs = (const float*)d_in[0];   // (B,3) float32
    const float* emb    = (const float*)d_in[1];   // (7114752,2) float32
    float* out = (float*)d_out;                    // (B,32) float32

    const int npoints = in_sizes[0] / 3;           // B = 1<<20
    const int total   = npoints * 16;              // one thread per (point, level)
    const int blocks  = (total + 255) / 256;
    hipLaunchKernelGGL(hashenc_fwd, dim3(blocks), dim3(256), 0, stream,
                       inputs, emb, out, npoints);
}